// StateSpaceModel_67731634258405
// MI455X (gfx1250) — compile-verified
//
#include <hip/hip_runtime.h>
#include <math.h>

// dims from reference
#define T_STEPS 128
#define BATCH   256
#define ADIM    16
#define ZDIM    32
#define KMIX    16
#define OUT_STRIDE (2*ZDIM + 2*ZDIM*ZDIM)   // 2112 floats per (t,b)

typedef __attribute__((ext_vector_type(2))) float v2f;
typedef __attribute__((ext_vector_type(8))) float v8f;

// ---------- WMMA f32 16x16x4 fragment helpers (ISA 7.12.2 layouts) ----------
// A (16x4, MxK): lanes 0-15 M=0..15 / K={k0,k0+1}; lanes 16-31 K={k0+2,k0+3}
__device__ __forceinline__ v2f frag_a(const float* m, int ld, int row0, int k0, int lane) {
    int half = lane >> 4, r = lane & 15;
    v2f a;
    a.x = m[(row0 + r) * ld + k0 + 2 * half + 0];
    a.y = m[(row0 + r) * ld + k0 + 2 * half + 1];
    return a;
}
// A = srcT : A[mrow][k] = src[k][mrow]
__device__ __forceinline__ v2f frag_aT(const float* m, int ld, int row0, int k0, int lane) {
    int half = lane >> 4, r = lane & 15;
    v2f a;
    a.x = m[(k0 + 2 * half + 0) * ld + row0 + r];
    a.y = m[(k0 + 2 * half + 1) * ld + row0 + r];
    return a;
}
// B (4x16, KxN): lanes 0-15 N=0..15 / K={k0,k0+1}; lanes 16-31 K={k0+2,k0+3}
__device__ __forceinline__ v2f frag_b(const float* m, int ld, int k0, int col0, int lane) {
    int half = lane >> 4, n = lane & 15;
    v2f b;
    b.x = m[(k0 + 2 * half + 0) * ld + col0 + n];
    b.y = m[(k0 + 2 * half + 1) * ld + col0 + n];
    return b;
}
// B = srcT : B[k][n] = src[n][k]
__device__ __forceinline__ v2f frag_bT(const float* m, int ld, int k0, int col0, int lane) {
    int half = lane >> 4, n = lane & 15;
    v2f b;
    b.x = m[(col0 + n) * ld + k0 + 2 * half + 0];
    b.y = m[(col0 + n) * ld + k0 + 2 * half + 1];
    return b;
}
// C/D (16x16): VGPR r -> M = r + 8*half, N = lane&15
__device__ __forceinline__ v8f frag_c(const float* m, int ld, int row0, int col0, int lane) {
    int half = lane >> 4, n = lane & 15;
    v8f c;
#pragma unroll
    for (int r = 0; r < 8; ++r) c[r] = m[(row0 + r + 8 * half) * ld + col0 + n];
    return c;
}
__device__ __forceinline__ void store_c(float* m, int ld, int row0, int col0, int lane, v8f c) {
    int half = lane >> 4, n = lane & 15;
#pragma unroll
    for (int r = 0; r < 8; ++r) m[(row0 + r + 8 * half) * ld + col0 + n] = c[r];
}
__device__ __forceinline__ v8f vzero8() {
    v8f z;
#pragma unroll
    for (int i = 0; i < 8; ++i) z[i] = 0.f;
    return z;
}
__device__ __forceinline__ v8f wmma4(v2f a, v2f b, v8f c) {
    // 8 args: (neg_a, A, neg_b, B, c_mod, C, reuse_a, reuse_b)
    return __builtin_amdgcn_wmma_f32_16x16x4_f32(false, a, false, b, (short)0, c, false, false);
}

__device__ __forceinline__ float sigmoidf_(float x) { return 1.0f / (1.0f + __expf(-x)); }

// per-wave LDS working-set size in floats
#define WAVE_WSZ (1024 /*P*/ + 1024 /*U*/ + 512 /*M*/ + 768 /*Aug16x48*/ + 1024 /*At*/ + 512 /*Ct*/ \
                  + 16 /*h*/ + 16 /*c*/ + 64 /*gates*/ + 16 /*wt*/ + 32 /*meanp*/ + 32 /*meant*/ \
                  + 32 /*nmean*/ + 16 /*rv*/)
#define SHARED_FLOATS (16384 + 8192 + 1024 + 256 + 1024 + 1024 + 64 + 2 * WAVE_WSZ)

__global__ void kalman_lstm_fused(const float* __restrict__ as_,
                                  const float* __restrict__ A_K,   // [16,32,32]
                                  const float* __restrict__ C_K,   // [16,16,32]
                                  const float* __restrict__ Q_L,   // [32,32]
                                  const float* __restrict__ R_L,   // [16,16]
                                  const float* __restrict__ W_ih,  // [64,16]
                                  const float* __restrict__ W_hh,  // [64,16]
                                  const float* __restrict__ b_ih,  // [64]
                                  const float* __restrict__ b_hh,  // [64]
                                  float* __restrict__ out)         // [T,B,2112]
{
    extern __shared__ float smem[];
    const int tid  = threadIdx.x;       // 0..63
    const int wave = tid >> 5;          // 0..1
    const int lane = tid & 31;
    const int b    = blockIdx.x * 2 + wave;

    // ---- shared (block-wide) constants ----
    float* sAK   = smem;                // 16384
    float* sCK   = sAK + 16384;         // 8192
    float* sQ    = sCK + 8192;          // 1024
    float* sR    = sQ + 1024;           // 256
    float* sWih  = sR + 256;            // 1024
    float* sWhh  = sWih + 1024;         // 1024
    float* sBias = sWhh + 1024;         // 64
    // ---- per-wave working set ----
    float* Wb    = sBias + 64 + wave * WAVE_WSZ;
    float* P     = Wb;                  // 32x32 cov (also cov_t / next_cov)
    float* U     = P + 1024;            // 32x32 temp
    float* M     = U + 1024;            // 16x32  C*P
    float* Aug   = M + 512;             // 16x48  [S | Y]
    float* At    = Aug + 768;           // 32x32 mixed A_t
    float* Ct    = At + 1024;           // 16x32 mixed C_t
    float* hS    = Ct + 512;            // 16 lstm h
    float* cS    = hS + 16;             // 16 lstm c
    float* gS    = cS + 16;             // 64 gates
    float* wt    = gS + 64;             // 16 softmax weights
    float* meanp = wt + 16;             // 32
    float* meant = meanp + 32;          // 32
    float* nmean = meant + 32;          // 32
    float* rv    = nmean + 32;          // 16 innovation

    // ================= preamble: fill shared constants =================
    for (int i = tid; i < 16384; i += 64) sAK[i] = A_K[i];
    for (int i = tid; i < 8192;  i += 64) sCK[i] = C_K[i];
    for (int i = tid; i < 1024;  i += 64) { sWih[i] = W_ih[i]; sWhh[i] = W_hh[i]; }
    sBias[tid] = b_ih[tid] + b_hh[tid];
    for (int i = tid; i < 1024; i += 64) {             // Q = Q_L Q_L^T + 1e-3 I
        int r = i >> 5, cc = i & 31;
        float acc = (r == cc) ? 1e-3f : 0.f;
        for (int k = 0; k < 32; ++k) acc += Q_L[r * 32 + k] * Q_L[cc * 32 + k];
        sQ[i] = acc;
    }
    for (int i = tid; i < 256; i += 64) {              // R = R_L R_L^T + 1e-3 I
        int r = i >> 4, cc = i & 15;
        float acc = (r == cc) ? 1e-3f : 0.f;
        for (int k = 0; k < 16; ++k) acc += R_L[r * 16 + k] * R_L[cc * 16 + k];
        sR[i] = acc;
    }
    // per-wave init: P = I (INITIAL_STD^2 = 1), mean = 0, lstm state = 0
    for (int i = lane; i < 1024; i += 32) { int r = i >> 5, cc = i & 31; P[i] = (r == cc) ? 1.0f : 0.0f; }
    meanp[lane] = 0.f;
    if (lane < 16) { hS[lane] = 0.f; cS[lane] = 0.f; }
    __syncthreads();

    const int gjrow = lane & 15;        // Gauss-Jordan row
    const int ch    = lane >> 4;        // column half: [ch*24, ch*24+24)

    for (int t = 0; t < T_STEPS; ++t) {
        const float* x_t = as_ + ((size_t)t * BATCH + b) * ADIM;

        // -------- LSTM step: gates = x W_ih^T + h W_hh^T + bias --------
#pragma unroll
        for (int gi = 0; gi < 2; ++gi) {
            int g = lane + gi * 32;
            float acc = sBias[g];
#pragma unroll
            for (int k = 0; k < 16; ++k) acc += x_t[k] * sWih[g * 16 + k];
#pragma unroll
            for (int k = 0; k < 16; ++k) acc += hS[k] * sWhh[g * 16 + k];
            gS[g] = acc;
        }
        __syncthreads();
        if (lane < 16) {
            float ig = sigmoidf_(gS[lane]);
            float fg = sigmoidf_(gS[lane + 16]);
            float gg = tanhf(gS[lane + 32]);
            float og = sigmoidf_(gS[lane + 48]);
            float cn = fg * cS[lane] + ig * gg;
            cS[lane] = cn;
            hS[lane] = og * tanhf(cn);
        }
        __syncthreads();
        if (lane < 16) {                 // softmax over 16 (redundant per-lane reduce, cheap)
            float mx = hS[0];
#pragma unroll
            for (int k = 1; k < 16; ++k) mx = fmaxf(mx, hS[k]);
            float s = 0.f;
#pragma unroll
            for (int k = 0; k < 16; ++k) s += __expf(hS[k] - mx);
            wt[lane] = __expf(hS[lane] - mx) / s;
        }
        __syncthreads();

        // -------- mix A_t, C_t from the 16 candidates --------
        float wreg[16];
#pragma unroll
        for (int k = 0; k < 16; ++k) wreg[k] = wt[k];
        for (int e = lane; e < 1024; e += 32) {
            float acc = 0.f;
#pragma unroll
            for (int k = 0; k < 16; ++k) acc += wreg[k] * sAK[k * 1024 + e];
            At[e] = acc;
        }
        for (int e = lane; e < 512; e += 32) {
            float acc = 0.f;
#pragma unroll
            for (int k = 0; k < 16; ++k) acc += wreg[k] * sCK[k * 512 + e];
            Ct[e] = acc;
        }
        __syncthreads();

        // -------- M = C_t (16x32) @ P (32x32) --------
#pragma unroll
        for (int j2 = 0; j2 < 2; ++j2) {
            v8f c8 = vzero8();
#pragma unroll
            for (int kk = 0; kk < 8; ++kk)
                c8 = wmma4(frag_a(Ct, 32, 0, kk * 4, lane), frag_b(P, 32, kk * 4, j2 * 16, lane), c8);
            store_c(M, 32, 0, j2 * 16, lane, c8);
        }
        __syncthreads();

        // -------- S = M @ C_t^T + R  -> Aug[:,0:16];  Aug[:,16:48] = M --------
        {
            v8f c8 = frag_c(sR, 16, 0, 0, lane);
#pragma unroll
            for (int kk = 0; kk < 8; ++kk)
                c8 = wmma4(frag_a(M, 32, 0, kk * 4, lane), frag_bT(Ct, 32, kk * 4, 0, lane), c8);
            store_c(Aug, 48, 0, 0, lane, c8);
        }
        for (int e = lane; e < 512; e += 32) {
            int r = e >> 5, cc = e & 31;
            Aug[r * 48 + 16 + cc] = M[e];
        }
        __syncthreads();

        // -------- Gauss-Jordan: [S | M] -> [I | Y],  Y = S^-1 M  (K = Y^T) --------
        for (int p = 0; p < 16; ++p) {
            float ip = 1.0f / Aug[p * 48 + p];
            if (gjrow == p) {
                for (int cc = ch * 24; cc < ch * 24 + 24; ++cc) Aug[p * 48 + cc] *= ip;
            }
            __syncthreads();
            float f = Aug[gjrow * 48 + p];
            __syncthreads();
            if (gjrow != p) {
                for (int cc = ch * 24; cc < ch * 24 + 24; ++cc)
                    Aug[gjrow * 48 + cc] -= f * Aug[p * 48 + cc];
            }
            __syncthreads();
        }

        // -------- mean updates (vector work, 32 lanes) --------
        if (lane < 16) {                                 // r = a_t - C_t meanp
            float acc = x_t[lane];
#pragma unroll
            for (int z = 0; z < 32; ++z) acc -= Ct[lane * 32 + z] * meanp[z];
            rv[lane] = acc;
        }
        __syncthreads();
        {                                                // mean_t = meanp + Y^T r
            float acc = meanp[lane];
#pragma unroll
            for (int a2 = 0; a2 < 16; ++a2) acc += Aug[a2 * 48 + 16 + lane] * rv[a2];
            meant[lane] = acc;
        }
        __syncthreads();
        {                                                // next_mean = A_t mean_t
            float acc = 0.f;
#pragma unroll
            for (int z = 0; z < 32; ++z) acc += At[lane * 32 + z] * meant[z];
            nmean[lane] = acc;
        }
        __syncthreads();

        // -------- cov_t = P - Y^T @ M  (in place into P; f32 WMMA has no A-neg -> negate frag) --------
#pragma unroll
        for (int i2 = 0; i2 < 2; ++i2)
#pragma unroll
            for (int j2 = 0; j2 < 2; ++j2) {
                v8f c8 = frag_c(P, 32, i2 * 16, j2 * 16, lane);
#pragma unroll
                for (int kk = 0; kk < 4; ++kk) {
                    v2f a = frag_aT(Aug + 16, 48, i2 * 16, kk * 4, lane);  // Y^T
                    a.x = -a.x; a.y = -a.y;
                    c8 = wmma4(a, frag_b(M, 32, kk * 4, j2 * 16, lane), c8);
                }
                store_c(P, 32, i2 * 16, j2 * 16, lane, c8);
            }
        __syncthreads();

        // -------- emit mean_t, cov_t, next_mean --------
        float* outp = out + ((size_t)t * BATCH + b) * OUT_STRIDE;
        outp[lane] = meant[lane];
        for (int e = lane; e < 1024; e += 32) outp[32 + e] = P[e];
        outp[32 + 1024 + lane] = nmean[lane];

        // -------- U = A_t @ cov_t --------
#pragma unroll
        for (int i2 = 0; i2 < 2; ++i2)
#pragma unroll
            for (int j2 = 0; j2 < 2; ++j2) {
                v8f c8 = vzero8();
#pragma unroll
                for (int kk = 0; kk < 8; ++kk)
                    c8 = wmma4(frag_a(At, 32, i2 * 16, kk * 4, lane),
                               frag_b(P, 32, kk * 4, j2 * 16, lane), c8);
                store_c(U, 32, i2 * 16, j2 * 16, lane, c8);
            }
        __syncthreads();

        // -------- next_cov = U @ A_t^T + Q  (into P for next iteration) --------
#pragma unroll
        for (int i2 = 0; i2 < 2; ++i2)
#pragma unroll
            for (int j2 = 0; j2 < 2; ++j2) {
                v8f c8 = frag_c(sQ, 32, i2 * 16, j2 * 16, lane);
#pragma unroll
                for (int kk = 0; kk < 8; ++kk)
                    c8 = wmma4(frag_a(U, 32, i2 * 16, kk * 4, lane),
                               frag_bT(At, 32, kk * 4, j2 * 16, lane), c8);
                store_c(P, 32, i2 * 16, j2 * 16, lane, c8);
            }
        __syncthreads();

        // -------- emit next_cov; roll mean --------
        for (int e = lane; e < 1024; e += 32) outp[32 + 1024 + 32 + e] = P[e];
        meanp[lane] = nmean[lane];
        __syncthreads();
    }
}

extern "C" void kernel_launch(void* const* d_in, const int* in_sizes, int n_in,
                              void* d_out, int out_size, void* d_ws, size_t ws_size,
                              hipStream_t stream) {
    const float* as_  = (const float*)d_in[0];
    const float* A_K  = (const float*)d_in[1];
    const float* C_K  = (const float*)d_in[2];
    const float* Q_L  = (const float*)d_in[3];
    const float* R_L  = (const float*)d_in[4];
    const float* W_ih = (const float*)d_in[5];
    const float* W_hh = (const float*)d_in[6];
    const float* b_ih = (const float*)d_in[7];
    const float* b_hh = (const float*)d_in[8];
    float* out = (float*)d_out;

    const size_t smem_bytes = (size_t)SHARED_FLOATS * sizeof(float);  // ~149 KB (<320 KB/WGP)
    (void)hipFuncSetAttribute((const void*)kalman_lstm_fused,
                              hipFuncAttributeMaxDynamicSharedMemorySize, (int)smem_bytes);

    // 128 blocks x 64 threads (2 waves = 2 batch chains per block), B = 256 chains total
    kalman_lstm_fused<<<dim3(BATCH / 2), dim3(64), smem_bytes, stream>>>(
        as_, A_K, C_K, Q_L, R_L, W_ih, W_hh, b_ih, b_hh, out);
}